// ClusterNorm1dv3_37151467110659
// MI455X (gfx1250) — compile-verified
//
#include <hip/hip_runtime.h>

// ClusterNorm1d: out[b,d,k] = sum_e Sinv[k,d,e] * (x[b,e,k] - mu[e,k])
// B=8192, D=64, K=128.  Per-cluster 64x64 GEMM via V_WMMA_F32_16X16X4_F32.

typedef __attribute__((ext_vector_type(2))) float v2f;
typedef __attribute__((ext_vector_type(8))) float v8f;

#define NB      8192
#define ND      64
#define NK      128
#define TILE_K  8            // clusters per workgroup (one per wave)
#define TILE_B  64           // batch rows per workgroup
#define ROWPAD  68           // padded LDS row stride (floats) for 64-wide rows
#define MATSZ   (64 * ROWPAD)  // 4352 floats per padded 64x64 tile
#define OSTRIDE 520          // out-staging stride per b row (64*8 + 8 pad)

__global__ __launch_bounds__(256)
void clusternorm_wmma_f32(const float* __restrict__ x,
                          const float* __restrict__ mu,
                          const float* __restrict__ Sinv,
                          float* __restrict__ out)
{
  extern __shared__ float lds[];
  float* s_lds  = lds;                      // 8 * MATSZ floats (Sinv, padded)
  float* x_lds  = lds + TILE_K * MATSZ;     // 8 * MATSZ floats (xc, padded; reused as out staging)
  float* mu_lds = lds + 2 * TILE_K * MATSZ; // 64 * 8 floats

  const int tid = threadIdx.x;
  const int k0  = blockIdx.x * TILE_K;      // k-blocks fastest -> L2 line reuse across k0
  const int b0  = blockIdx.y * TILE_B;

  // ---- Issue ALL x-tile global loads up front (no LDS dependence) so they
  //      overlap the Sinv/mu staging below.  16 (b,e) pairs/thread, 2 float4 each.
  float4 xa[16], xb[16];
  #pragma unroll
  for (int i = 0; i < 16; ++i) {
    int p = tid + i * 256;              // (b,e) pair 0..4095
    int b = p >> 6;
    int e = p & 63;
    const float4* g = (const float4*)(x + ((size_t)(b0 + b) * ND + e) * NK + k0);
    xa[i] = g[0];
    xb[i] = g[1];
  }

  // ---- Stage Sinv[k0:k0+8] (contiguous 128KB) into padded LDS, coalesced float4
  {
    const float4* g = (const float4*)(Sinv + (size_t)k0 * ND * ND);
    #pragma unroll
    for (int i = 0; i < 32; ++i) {
      int gi  = tid + i * 256;          // float4 index 0..8191
      int row = gi >> 4;                // (k*64 + d), rows are exactly 16 float4s
      int c4  = gi & 15;
      float4 v = g[gi];
      *(float4*)(s_lds + row * ROWPAD + c4 * 4) = v;
    }
  }
  // ---- Stage mu[:, k0:k0+8] -> mu_lds[e][kk]
  if (tid < 64) {
    const float4* g = (const float4*)(mu + (size_t)tid * NK + k0);
    float4 a = g[0], b = g[1];
    *(float4*)(mu_lds + tid * 8)     = a;
    *(float4*)(mu_lds + tid * 8 + 4) = b;
  }
  __syncthreads();

  // ---- Subtract mu, store transposed into x_lds[k][b][e] (e contig, b stride ROWPAD)
  #pragma unroll
  for (int i = 0; i < 16; ++i) {
    int p = tid + i * 256;
    int b = p >> 6;
    int e = p & 63;
    float4 ma = *(const float4*)(mu_lds + e * 8);
    float4 mb = *(const float4*)(mu_lds + e * 8 + 4);
    float c[8] = { xa[i].x - ma.x, xa[i].y - ma.y, xa[i].z - ma.z, xa[i].w - ma.w,
                   xb[i].x - mb.x, xb[i].y - mb.y, xb[i].z - mb.z, xb[i].w - mb.w };
    #pragma unroll
    for (int kk = 0; kk < 8; ++kk)
      x_lds[kk * MATSZ + b * ROWPAD + e] = c[kk];
  }
  __syncthreads();

  // ---- Per-wave GEMM: wave w owns cluster k0+w.
  // out(d,b) = Sinv(d,e) @ xc(e,b): A = 16x4 of Sinv, B = 4x16 of xc.
  const int wave = tid >> 5;
  const int lane = tid & 31;
  const int lrow = lane & 15;           // M (=d) for A, N (=b) for B
  const int lhi  = lane >> 4;           // K offset = 2*lhi

  const float* Sb = s_lds + wave * MATSZ;
  const float* Xb = x_lds + wave * MATSZ;

  v8f acc[4][4];
  #pragma unroll
  for (int nd = 0; nd < 4; ++nd)
    #pragma unroll
    for (int nb = 0; nb < 4; ++nb)
      acc[nd][nb] = (v8f){};

  for (int e0 = 0; e0 < 64; e0 += 4) {
    v2f Af[4], Bf[4];
    #pragma unroll
    for (int nd = 0; nd < 4; ++nd)      // A: lane holds S[d = nd*16+lrow][e0+2*lhi +{0,1}]
      Af[nd] = *(const v2f*)(Sb + (nd * 16 + lrow) * ROWPAD + e0 + 2 * lhi);
    #pragma unroll
    for (int nb = 0; nb < 4; ++nb)      // B: lane holds xc[e0+2*lhi +{0,1}][b = nb*16+lrow]
      Bf[nb] = *(const v2f*)(Xb + (nb * 16 + lrow) * ROWPAD + e0 + 2 * lhi);
    #pragma unroll
    for (int nd = 0; nd < 4; ++nd)
      #pragma unroll
      for (int nb = 0; nb < 4; ++nb)
        acc[nd][nb] = __builtin_amdgcn_wmma_f32_16x16x4_f32(
            false, Af[nd], false, Bf[nb], (short)0, acc[nd][nb], false, false);
  }

  __syncthreads();   // all waves done reading x_lds; reuse it for output staging

  // ---- Stage C tiles: o_lds[b*OSTRIDE + d*8 + wave]
  float* o_lds = x_lds;
  #pragma unroll
  for (int nd = 0; nd < 4; ++nd)
    #pragma unroll
    for (int nb = 0; nb < 4; ++nb)
      #pragma unroll
      for (int v = 0; v < 8; ++v) {
        int d = nd * 16 + v + 8 * lhi;  // C/D layout: VGPR v -> M = v (+8 for hi lanes)
        int b = nb * 16 + lrow;
        o_lds[b * OSTRIDE + d * 8 + wave] = acc[nd][nb][v];
      }
  __syncthreads();

  // ---- Coalesced store: per (b,d) write 8 contiguous k floats (2x float4)
  for (int i = 0; i < 16; ++i) {
    int p = tid + i * 256;
    int b = p >> 6;
    int d = p & 63;
    float4 va = *(const float4*)(o_lds + b * OSTRIDE + d * 8);
    float4 vb = *(const float4*)(o_lds + b * OSTRIDE + d * 8 + 4);
    float4* g = (float4*)(out + ((size_t)(b0 + b) * ND + d) * NK + k0);
    g[0] = va;
    g[1] = vb;
  }
}

extern "C" void kernel_launch(void* const* d_in, const int* in_sizes, int n_in,
                              void* d_out, int out_size, void* d_ws, size_t ws_size,
                              hipStream_t stream) {
  const float* x    = (const float*)d_in[0];  // [B, D, K]
  const float* mu   = (const float*)d_in[1];  // [D, K]
  const float* Sinv = (const float*)d_in[2];  // [K, D, D]
  float* out        = (float*)d_out;          // [B, D, K]

  dim3 grid(NK / TILE_K, NB / TILE_B);        // (16, 128), k-blocks fastest
  size_t shmem = (size_t)(2 * TILE_K * MATSZ + 64 * 8) * sizeof(float); // ~274 KB
  clusternorm_wmma_f32<<<grid, 256, shmem, stream>>>(x, mu, Sinv, out);
}